// MambaLayer_47107201302863
// MI455X (gfx1250) — compile-verified
//
#include <hip/hip_runtime.h>
#include <hip/hip_bf16.h>
#include <math.h>

// ---------------------------------------------------------------------------
// Mamba layer forward for MI455X (gfx1250, wave32, WMMA + async-to-LDS).
//   BATCH=2, SEQ=1024, D_MODEL=1024, D_STATE=16, D_CONV=4, D_INNER=2048,
//   DT_RANK=64
// ---------------------------------------------------------------------------

#define BATCH    2
#define SEQ      1024
#define D_MODEL  1024
#define D_STATE  16
#define D_CONV   4
#define D_INNER  2048
#define DT_RANK  64
#define XPROJ    (DT_RANK + 2 * D_STATE)   // 96

typedef __attribute__((ext_vector_type(2))) float v2f;
typedef __attribute__((ext_vector_type(8))) float v8f;

// ---------------------------------------------------------------------------
// WMMA GEMM: C[M,N] = A[M,K] @ W[N,K]^T + bias[N], full f32 precision via
// V_WMMA_F32_16X16X4_F32. One wave computes a 16 x (16*NT) strip; the A
// fragment is shared across NT accumulators. N,K are compile-time so all
// W-tile addresses fold into one base + 24-bit immediate offsets, and loads
// for each 16-wide K chunk are issued as a batch ahead of the 4*NT wmmas so
// the scheduler can pipeline loads across chunks.
//
// ISA VGPR layouts (CDNA5 ISA):
//   A 16x4 f32 : a[v] = A[lane%16][2*(lane/16)+v]
//   B 4x16 f32 : b[v] = B[2*(lane/16)+v][lane%16] = W[lane%16][2*(lane/16)+v]
//   C/D 16x16  : acc[v] -> row = v + 8*(lane/16), col = lane%16
// ---------------------------------------------------------------------------
template <int NT, int N, int K>
__global__ void __launch_bounds__(32)
wmma_gemm_xwt(const float* __restrict__ A, const float* __restrict__ W,
              const float* __restrict__ bias, float* __restrict__ C) {
  const int lane = threadIdx.x & 31;
  constexpr int ntn = N / (16 * NT);
  const int tm  = (blockIdx.x / ntn) << 4;
  const int tn0 = (blockIdx.x % ntn) * (16 * NT);

  const int r  = lane & 15;          // row within A tile / row within W tile
  const int kk = (lane >> 4) << 1;   // K sub-offset {0,2}

  const float* __restrict__ Ar = A + (size_t)(tm + r) * K + kk;
  const float* __restrict__ Wr = W + (size_t)(tn0 + r) * K + kk;

  v8f acc[NT];
#pragma unroll
  for (int j = 0; j < NT; ++j) acc[j] = (v8f){};

  for (int k = 0; k < K; k += 16) {
    v2f a[4];
    v2f b[NT][4];
#pragma unroll
    for (int u = 0; u < 4; ++u) a[u] = *(const v2f*)(Ar + k + 4 * u);
#pragma unroll
    for (int j = 0; j < NT; ++j) {
#pragma unroll
      for (int u = 0; u < 4; ++u) {
        b[j][u] = *(const v2f*)(Wr + j * 16 * K + k + 4 * u);  // const offsets
      }
    }
#pragma unroll
    for (int u = 0; u < 4; ++u) {
#pragma unroll
      for (int j = 0; j < NT; ++j) {
        acc[j] = __builtin_amdgcn_wmma_f32_16x16x4_f32(
            false, a[u], false, b[j][u], (short)0, acc[j], false, false);
      }
    }
  }

  const int rb = tm + ((lane >> 4) << 3);
#pragma unroll
  for (int j = 0; j < NT; ++j) {
    const int   col  = tn0 + 16 * j + r;
    const float bcol = bias[col];
#pragma unroll
    for (int v = 0; v < 8; ++v) {
      C[(size_t)(rb + v) * N + col] = acc[j][v] + bcol;
    }
  }
}

// ---------------------------------------------------------------------------
// Depthwise causal conv1d (kernel=4, left pad 3) + bias; c fastest -> coalesced.
// ---------------------------------------------------------------------------
__global__ void __launch_bounds__(256)
conv1d_kernel(const float* __restrict__ xin, const float* __restrict__ w,
              const float* __restrict__ cb, float* __restrict__ xconv) {
  const int idx = blockIdx.x * 256 + threadIdx.x;
  const int c = idx & (D_INNER - 1);
  const int s = (idx >> 11) & (SEQ - 1);
  const int b = idx >> 21;
  float acc = cb[c];
#pragma unroll
  for (int k = 0; k < D_CONV; ++k) {
    const int sp = s - (D_CONV - 1) + k;
    if (sp >= 0) {
      acc = fmaf(xin[((size_t)(b * SEQ + sp)) * D_INNER + c], w[c * D_CONV + k], acc);
    }
  }
  xconv[idx] = acc;
}

// ---------------------------------------------------------------------------
// delta[b*SEQ+s] = softplus( dot(x_dbc[b,s,0:64], dt[0:64]) )
// ---------------------------------------------------------------------------
__global__ void __launch_bounds__(256)
delta_kernel(const float* __restrict__ xdbc, const float* __restrict__ dt,
             float* __restrict__ delta) {
  const int i = blockIdx.x * 256 + threadIdx.x;
  if (i >= BATCH * SEQ) return;
  const float* __restrict__ row = xdbc + (size_t)i * XPROJ;
  float s = 0.0f;
#pragma unroll 8
  for (int r = 0; r < DT_RANK; ++r) s = fmaf(row[r], dt[r], s);
  delta[i] = (s > 20.0f) ? s : log1pf(__expf(s));
}

// ---------------------------------------------------------------------------
// Selective scan. One lane per (batch, channel): h[16] and A[c,:] in VGPRs.
// delta/B/C tiles (shared across a batch) are DOUBLE-BUFFERED in LDS and
// streamed with GLOBAL_LOAD_ASYNC_TO_LDS_B128/B32 (ASYNCcnt), so the next
// 64-timestep tile loads while the current tile is scanned.
// ---------------------------------------------------------------------------
#define TT 64   // timesteps per LDS tile

__device__ __forceinline__ void async_ld_b128(unsigned lds_off, const void* gaddr) {
  asm volatile("global_load_async_to_lds_b128 %0, %1, off"
               :: "v"(lds_off), "v"((unsigned long long)(size_t)gaddr) : "memory");
}
__device__ __forceinline__ void async_ld_b32(unsigned lds_off, const void* gaddr) {
  asm volatile("global_load_async_to_lds_b32 %0, %1, off"
               :: "v"(lds_off), "v"((unsigned long long)(size_t)gaddr) : "memory");
}
__device__ __forceinline__ void wait_async0() {
  asm volatile("s_wait_asynccnt 0x0" ::: "memory");
}

__global__ void __launch_bounds__(256)
scan_kernel(const float* __restrict__ xin, const float* __restrict__ xdbc,
            const float* __restrict__ delta, const float* __restrict__ Amat,
            const float* __restrict__ Dp, float* __restrict__ y) {
  const int b = blockIdx.y;
  const int c = blockIdx.x * 256 + threadIdx.x;

  __shared__ __align__(16) float sD[2][TT];
  __shared__ __align__(16) float sB[2][TT][D_STATE];
  __shared__ __align__(16) float sC[2][TT][D_STATE];

  float h[D_STATE];
  float a[D_STATE];
#pragma unroll
  for (int n = 0; n < D_STATE; ++n) {
    h[n] = 0.0f;
    a[n] = Amat[c * D_STATE + n];
  }
  const float dpc = Dp[c];

  const size_t xin_base = (size_t)b * SEQ * D_INNER + c;
  const size_t dbc_base = (size_t)b * SEQ * XPROJ;

  // stage one 64-timestep tile of delta/B/C into LDS buffer `buf` (async)
  auto stage = [&](int buf, int t0) {
    const int t = threadIdx.x >> 2;        // 0..63
    const int q = threadIdx.x & 3;         // 0..3 -> 4 floats each
    const float* gB = xdbc + dbc_base + (size_t)(t0 + t) * XPROJ + DT_RANK + q * 4;
    async_ld_b128((unsigned)(size_t)&sB[buf][t][q * 4], gB);
    async_ld_b128((unsigned)(size_t)&sC[buf][t][q * 4], gB + D_STATE);
    if (threadIdx.x < TT) {
      async_ld_b32((unsigned)(size_t)&sD[buf][threadIdx.x],
                   delta + b * SEQ + t0 + threadIdx.x);
    }
  };

  stage(0, 0);
  wait_async0();
  __syncthreads();

  const int NTILE = SEQ / TT;
  for (int tile = 0; tile < NTILE; ++tile) {
    const int buf = tile & 1;
    if (tile + 1 < NTILE) stage(buf ^ 1, (tile + 1) * TT);   // overlap with compute

    const int t0 = tile * TT;
    for (int i = 0; i < TT; ++i) {
      const int t = t0 + i;
      const float xt = xin[xin_base + (size_t)t * D_INNER];
      const float d  = sD[buf][i];
      float acc = dpc * xt;
#pragma unroll
      for (int n = 0; n < D_STATE; ++n) {
        float z = d * a[n];
        z = fminf(fmaxf(z, -5.0f), 5.0f);
        const float decay = __expf(-z);          // v_exp_f32 (TRANS, co-exec)
        h[n] = fmaf(h[n], decay, sB[buf][i][n] * xt);
        acc  = fmaf(h[n], sC[buf][i][n], acc);
      }
      y[xin_base + (size_t)t * D_INNER] = acc;
    }

    wait_async0();       // next tile's async loads have landed in LDS
    __syncthreads();     // all waves done with current buffer
  }
}

// ---------------------------------------------------------------------------
// Launch
// ---------------------------------------------------------------------------
extern "C" void kernel_launch(void* const* d_in, const int* in_sizes, int n_in,
                              void* d_out, int out_size, void* d_ws, size_t ws_size,
                              hipStream_t stream) {
  const float* x      = (const float*)d_in[0];
  const float* in_w   = (const float*)d_in[1];
  const float* in_b   = (const float*)d_in[2];
  const float* conv_w = (const float*)d_in[3];
  const float* conv_b = (const float*)d_in[4];
  const float* xp_w   = (const float*)d_in[5];
  const float* xp_b   = (const float*)d_in[6];
  const float* Amat   = (const float*)d_in[7];
  const float* Dp     = (const float*)d_in[8];
  const float* dt     = (const float*)d_in[9];
  const float* out_w  = (const float*)d_in[10];
  const float* out_b  = (const float*)d_in[11];
  float* out = (float*)d_out;

  // Workspace carve-up (floats)
  float* ws     = (float*)d_ws;
  float* x_in   = ws;                                       // 4194304
  float* x_conv = x_in   + (size_t)BATCH * SEQ * D_INNER;   // 4194304
  float* x_dbc  = x_conv + (size_t)BATCH * SEQ * D_INNER;   // 196608
  float* delta  = x_dbc  + (size_t)BATCH * SEQ * XPROJ;     // 2048
  float* ybuf   = delta  + (size_t)BATCH * SEQ;             // 4194304

  const int M = BATCH * SEQ;  // 2048

  // 1) x_in = x @ in_w^T + in_b     (M=2048, N=2048, K=1024), 16x64 per wave
  wmma_gemm_xwt<4, D_INNER, D_MODEL><<<(M / 16) * (D_INNER / 64), 32, 0, stream>>>(
      x, in_w, in_b, x_in);

  // 2) depthwise causal conv + bias
  conv1d_kernel<<<(BATCH * SEQ * D_INNER) / 256, 256, 0, stream>>>(
      x_in, conv_w, conv_b, x_conv);

  // 3) x_dbc = x_conv @ xp_w^T + xp_b   (M=2048, N=96, K=2048), 16x48 per wave
  wmma_gemm_xwt<3, XPROJ, D_INNER><<<(M / 16) * (XPROJ / 48), 32, 0, stream>>>(
      x_conv, xp_w, xp_b, x_dbc);

  // 4) delta = softplus(x_dbc[:,:64] @ dt)
  delta_kernel<<<(BATCH * SEQ + 255) / 256, 256, 0, stream>>>(x_dbc, dt, delta);

  // 5) selective scan -> ybuf (async double-buffered LDS staging)
  dim3 sgrid(D_INNER / 256, BATCH);
  scan_kernel<<<sgrid, 256, 0, stream>>>(x_in, x_dbc, delta, Amat, Dp, ybuf);

  // 6) out = ybuf @ out_w^T + out_b    (M=2048, N=1024, K=2048), 16x64 per wave
  wmma_gemm_xwt<4, D_MODEL, D_INNER><<<(M / 16) * (D_MODEL / 64), 32, 0, stream>>>(
      ybuf, out_w, out_b, out);
}